// NeighborhoodAggregationEmbedding_39745627357349
// MI455X (gfx1250) — compile-verified
//
#include <hip/hip_runtime.h>
#include <hip/hip_bf16.h>
#include <math.h>

#define BB 4
#define NN 512
#define EE 128
#define HH 4
#define DD 32
#define PP 6

typedef __attribute__((ext_vector_type(2))) float v2f;
typedef __attribute__((ext_vector_type(8))) float v8f;

// ---------------------------------------------------------------------------
// Kernel 0: fold query into kv_w/kv_b:  Wqk[p][h], Bqk[h]  (28 floats to ws)
// ---------------------------------------------------------------------------
__global__ void nae_prep(const float* __restrict__ kv_w, const float* __restrict__ kv_b,
                         const float* __restrict__ query, float* __restrict__ wqkb) {
    const int t = threadIdx.x;
    const float inv_sqrt_d = 0.17677669529663687f; // 1/sqrt(32)
    if (t < PP * HH) {
        int p = t / HH, h = t % HH;
        float s = 0.f;
        #pragma unroll
        for (int d = 0; d < DD; ++d)
            s = fmaf(query[h * DD + d], kv_w[p * (2 * EE) + h * DD + d], s);
        wqkb[p * HH + h] = s * inv_sqrt_d;
    } else if (t < PP * HH + HH) {
        int h = t - PP * HH;
        float s = 0.f;
        #pragma unroll
        for (int d = 0; d < DD; ++d)
            s = fmaf(query[h * DD + d], kv_b[h * DD + d], s);
        wqkb[PP * HH + h] = s * inv_sqrt_d;
    }
}

// ---------------------------------------------------------------------------
// Kernel 1: fused feats + logits + online softmax + weighted moments + ctx
// One wave32 per (b,n) row; 8 waves per block.
// ---------------------------------------------------------------------------
__global__ __launch_bounds__(256) void nae_attn(
    const float* __restrict__ positions, const unsigned char* __restrict__ pad,
    const float* __restrict__ kv_w, const float* __restrict__ kv_b,
    const float* __restrict__ wqkb, float* __restrict__ ctx) {

    const int lane = threadIdx.x & 31;
    const int wave = threadIdx.x >> 5;
    const int row  = blockIdx.x * 8 + wave;   // 0..2047
    const int b    = row >> 9;                 // / 512
    const int n    = row & (NN - 1);

    float Wqk[PP][HH], Bqk[HH];
    #pragma unroll
    for (int p = 0; p < PP; ++p)
        #pragma unroll
        for (int h = 0; h < HH; ++h) Wqk[p][h] = wqkb[p * HH + h];
    #pragma unroll
    for (int h = 0; h < HH; ++h) Bqk[h] = wqkb[PP * HH + h];

    const float px = positions[(b * NN + n) * 2 + 0];
    const float py = positions[(b * NN + n) * 2 + 1];

    float mx[HH], den[HH], S[HH][PP];
    #pragma unroll
    for (int h = 0; h < HH; ++h) {
        mx[h] = -3.0e38f; den[h] = 0.f;
        #pragma unroll
        for (int p = 0; p < PP; ++p) S[h][p] = 0.f;
    }

    for (int i = 0; i < NN / 32; ++i) {
        const int m = lane + 32 * i;
        const float qx = positions[(b * NN + m) * 2 + 0];
        const float qy = positions[(b * NN + m) * 2 + 1];
        const float dx = qx - px, dy = qy - py;
        const float r2 = dx * dx + dy * dy;
        const float dist = sqrtf(r2 + 1e-8f);
        const float hyp  = sqrtf(r2);
        float c = 1.f, s = 0.f;
        if (hyp > 0.f) { const float ir = 1.f / hyp; c = dx * ir; s = dy * ir; }
        const float idd = 1.f / (dist + 1e-8f);
        float f[PP];
        f[0] = dist; f[1] = c; f[2] = s; f[3] = dx * idd; f[4] = dy * idd; f[5] = log1pf(dist);
        const bool masked = (m == n) || (pad[b * NN + m] != 0);

        #pragma unroll
        for (int h = 0; h < HH; ++h) {
            float l = Bqk[h];
            #pragma unroll
            for (int p = 0; p < PP; ++p) l = fmaf(f[p], Wqk[p][h], l);
            if (masked) l = -3.0e38f;
            const float nm    = fmaxf(mx[h], l);
            const float scale = __expf(mx[h] - nm);
            const float e     = __expf(l - nm);
            den[h] = den[h] * scale + e;
            #pragma unroll
            for (int p = 0; p < PP; ++p) S[h][p] = fmaf(S[h][p], scale, e * f[p]);
            mx[h] = nm;
        }
    }

    // merge online-softmax partials across the 32 lanes of the wave
    #pragma unroll
    for (int off = 16; off > 0; off >>= 1) {
        #pragma unroll
        for (int h = 0; h < HH; ++h) {
            const float mo = __shfl_xor(mx[h], off, 32);
            const float dn = __shfl_xor(den[h], off, 32);
            float So[PP];
            #pragma unroll
            for (int p = 0; p < PP; ++p) So[p] = __shfl_xor(S[h][p], off, 32);
            const float nm = fmaxf(mx[h], mo);
            const float s1 = __expf(mx[h] - nm);
            const float s2 = __expf(mo - nm);
            den[h] = den[h] * s1 + dn * s2;
            #pragma unroll
            for (int p = 0; p < PP; ++p) S[h][p] = S[h][p] * s1 + So[p] * s2;
            mx[h] = nm;
        }
    }

    float invden[HH];
    #pragma unroll
    for (int h = 0; h < HH; ++h) invden[h] = 1.f / den[h];

    // ctx[b,n,e] = sum_p (S[h][p]/den[h]) * Wv[p][e] + bv[e],  h = e/32
    #pragma unroll
    for (int j = 0; j < HH; ++j) {
        const int e = 32 * j + lane;
        float acc = kv_b[EE + e];
        #pragma unroll
        for (int p = 0; p < PP; ++p)
            acc = fmaf(S[j][p] * invden[j], kv_w[p * (2 * EE) + EE + e], acc);
        ctx[row * EE + e] = acc;
    }
}

// ---------------------------------------------------------------------------
// Kernel 2: fused MLP on 16-row tiles:  GEMM(w1)+b1 -> LN -> GELU -> GEMM(w2)+b2
// 8 waves/block; wave w owns output columns [16w,16w+16); 32 WMMA k-steps each.
// ---------------------------------------------------------------------------
__global__ __launch_bounds__(256) void nae_mlp(
    const float* __restrict__ ctx,
    const float* __restrict__ w1, const float* __restrict__ b1,
    const float* __restrict__ ln_g, const float* __restrict__ ln_b,
    const float* __restrict__ w2, const float* __restrict__ b2,
    float* __restrict__ out) {

    __shared__ float xs[16 * 132];
    __shared__ float hs[16 * 132];

    const int tid  = threadIdx.x;
    const int lane = tid & 31;
    const int wave = tid >> 5;
    const int r0   = blockIdx.x * 16;

    // stage 16x128 ctx tile into LDS
    for (int i = tid; i < 16 * EE; i += 256) {
        const int rr = i >> 7, cc = i & (EE - 1);
        xs[rr * 132 + cc] = ctx[(r0 + rr) * EE + cc];
    }
    __syncthreads();

    const int mrow  = lane & 15;
    const int khalf = (lane < 16) ? 0 : 2;          // A/B K split per ISA layout
    const int col   = wave * 16 + (lane & 15);
    const int rowoff = (lane < 16) ? 0 : 8;         // C/D M split per ISA layout

    // ---- GEMM1: H = X @ w1 + b1 ----
    v8f acc;
    {
        const float bv = b1[col];
        #pragma unroll
        for (int r = 0; r < 8; ++r) acc[r] = bv;
    }
    #pragma unroll 4
    for (int kk = 0; kk < EE / 4; ++kk) {
        const int kb = kk * 4 + khalf;
        v2f a, bm;
        a[0]  = xs[mrow * 132 + kb + 0];
        a[1]  = xs[mrow * 132 + kb + 1];
        bm[0] = w1[(kb + 0) * EE + col];
        bm[1] = w1[(kb + 1) * EE + col];
        acc = __builtin_amdgcn_wmma_f32_16x16x4_f32(false, a, false, bm,
                                                    (short)0, acc, false, false);
    }
    #pragma unroll
    for (int r = 0; r < 8; ++r) hs[(r + rowoff) * 132 + col] = acc[r];
    __syncthreads();

    // ---- LayerNorm + exact GELU (each wave handles 2 rows) ----
    #pragma unroll
    for (int rr = wave * 2; rr < wave * 2 + 2; ++rr) {
        float xv[4];
        float sum = 0.f, sq = 0.f;
        #pragma unroll
        for (int j = 0; j < 4; ++j) {
            xv[j] = hs[rr * 132 + 32 * j + lane];
            sum += xv[j];
            sq  += xv[j] * xv[j];
        }
        #pragma unroll
        for (int off = 16; off > 0; off >>= 1) {
            sum += __shfl_xor(sum, off, 32);
            sq  += __shfl_xor(sq, off, 32);
        }
        const float mu  = sum * (1.f / (float)EE);
        const float var = sq * (1.f / (float)EE) - mu * mu;
        const float inv = rsqrtf(var + 1e-5f);
        #pragma unroll
        for (int j = 0; j < 4; ++j) {
            const int e = 32 * j + lane;
            const float y = (xv[j] - mu) * inv * ln_g[e] + ln_b[e];
            const float g = 0.5f * y * (1.f + erff(y * 0.70710678118654752f));
            hs[rr * 132 + e] = g;
        }
    }
    __syncthreads();

    // ---- GEMM2: out = G @ w2 + b2 ----
    v8f acc2;
    {
        const float bv = b2[col];
        #pragma unroll
        for (int r = 0; r < 8; ++r) acc2[r] = bv;
    }
    #pragma unroll 4
    for (int kk = 0; kk < EE / 4; ++kk) {
        const int kb = kk * 4 + khalf;
        v2f a, bm;
        a[0]  = hs[mrow * 132 + kb + 0];
        a[1]  = hs[mrow * 132 + kb + 1];
        bm[0] = w2[(kb + 0) * EE + col];
        bm[1] = w2[(kb + 1) * EE + col];
        acc2 = __builtin_amdgcn_wmma_f32_16x16x4_f32(false, a, false, bm,
                                                     (short)0, acc2, false, false);
    }
    #pragma unroll
    for (int r = 0; r < 8; ++r)
        out[(r0 + r + rowoff) * EE + col] = acc2[r];
}

// ---------------------------------------------------------------------------
extern "C" void kernel_launch(void* const* d_in, const int* in_sizes, int n_in,
                              void* d_out, int out_size, void* d_ws, size_t ws_size,
                              hipStream_t stream) {
    const float*         positions = (const float*)d_in[0];
    const unsigned char* pad       = (const unsigned char*)d_in[1];
    const float*         kv_w      = (const float*)d_in[2];
    const float*         kv_b      = (const float*)d_in[3];
    const float*         query     = (const float*)d_in[4];
    const float*         w1        = (const float*)d_in[5];
    const float*         b1        = (const float*)d_in[6];
    const float*         ln_g      = (const float*)d_in[7];
    const float*         ln_b      = (const float*)d_in[8];
    const float*         w2        = (const float*)d_in[9];
    const float*         b2        = (const float*)d_in[10];
    float* out = (float*)d_out;

    float* wqkb = (float*)d_ws;          // 28 floats
    float* ctx  = (float*)d_ws + 64;     // 2048*128 floats = 1 MB

    nae_prep<<<1, 32, 0, stream>>>(kv_w, kv_b, query, wqkb);
    nae_attn<<<(BB * NN) / 8, 256, 0, stream>>>(positions, pad, kv_w, kv_b, wqkb, ctx);
    nae_mlp<<<(BB * NN) / 16, 256, 0, stream>>>(ctx, w1, b1, ln_g, ln_b, w2, b2, out);
}